// Model_3470333575383
// MI455X (gfx1250) — compile-verified
//
#include <hip/hip_runtime.h>

// Problem constants (match reference).
#define N_EXPERTS     64
#define TOTAL_TOKENS  67108864   // 64M entries -> 256MB output stream
#define SEARCH_PAD    128        // LDS table padded with INT_MAX sentinels

// Native vector types so __builtin_nontemporal_{load,store} accept them
// (HIP float4/int4 are class types; clang's builtins need real vectors).
typedef __attribute__((ext_vector_type(4))) float v4f;
typedef __attribute__((ext_vector_type(4))) int   v4i;

// ---------------------------------------------------------------------------
// Kernel A: 64-element exclusive scan of per-expert counts.
//   ws[0..63]  = starts (int32, consumed by fill kernel)
//   ws[64]     = total dispatched tokens
//   out_starts = float(starts)  (second output of the reference tuple)
// Runs once with 64 threads; cost is noise next to the 256MB stream.
// ---------------------------------------------------------------------------
__global__ __launch_bounds__(N_EXPERTS)
void expert_scan_kernel(const int* __restrict__ counts,
                        int* __restrict__ ws,
                        float* __restrict__ out_starts) {
    __shared__ int s[N_EXPERTS];
    const int t = threadIdx.x;
    const int c = counts[t];
    s[t] = c;
    __syncthreads();
    // Hillis-Steele inclusive scan over 64 elements (6 steps).
#pragma unroll
    for (int off = 1; off < N_EXPERTS; off <<= 1) {
        const int v = (t >= off) ? s[t - off] : 0;
        __syncthreads();
        s[t] += v;
        __syncthreads();
    }
    const int incl  = s[t];
    const int start = incl - c;          // exclusive cumsum
    ws[t]         = start;
    out_starts[t] = (float)start;
    if (t == N_EXPERTS - 1) ws[N_EXPERTS] = incl;   // total
}

// ---------------------------------------------------------------------------
// Kernel B: fill m_out. One 4-token vector (16B) per thread so each wave32
// b128 store covers 512 contiguous bytes (max-bandwidth pattern); stores use
// the NT temporal hint since the 256MB stream exceeds the 192MB L2 and is
// never re-read. Expert lookup = branch-free 6-step upper-bound search in
// LDS; table padded to 128 entries with INT_MAX so no bounds checks, and
// equal starts (zero-count experts) resolve to the LAST expert, matching
// searchsorted(..., side='right') - 1. Lanes of a wave almost always probe
// the same entries -> LDS same-address broadcast, no bank conflicts.
// ---------------------------------------------------------------------------
__global__ __launch_bounds__(256)
void fill_m_indices_kernel(const int* __restrict__ ws,
                           const int* __restrict__ m_indices,
                           float* __restrict__ out) {
    __shared__ int s_starts[SEARCH_PAD];
    __shared__ int s_total;

    const int tid = threadIdx.x;
    if (tid < SEARCH_PAD) {
        s_starts[tid] = (tid < N_EXPERTS) ? ws[tid] : 0x7fffffff;
    }
    if (tid == 0) s_total = ws[N_EXPERTS];
    __syncthreads();

    const int total = s_total;
    const int base  = 4 * (int)(blockIdx.x * blockDim.x + threadIdx.x); // < 2^26

    v4f v;
    if (base + 3 < total) {
        // Fast path: all 4 tokens dispatched. One search, then a hoisted
        // boundary check per token (+1/+2/+3): common case is 3 compares
        // with never-taken branches (boundary inside a quad ~ 4/1M odds).
        int e = 0;
#pragma unroll
        for (int step = 32; step >= 1; step >>= 1) {
            if (s_starts[e + step] <= base) e += step;   // ds_load broadcast
        }
        int nxt = s_starts[e + 1];
        const int e0 = e;
        if (nxt <= base + 1) { do { ++e; nxt = s_starts[e + 1]; } while (nxt <= base + 1); }
        const int e1 = e;
        if (nxt <= base + 2) { do { ++e; nxt = s_starts[e + 1]; } while (nxt <= base + 2); }
        const int e2 = e;
        if (nxt <= base + 3) { do { ++e; nxt = s_starts[e + 1]; } while (nxt <= base + 3); }
        const int e3 = e;
        v = (v4f){(float)e0, (float)e1, (float)e2, (float)e3};
    } else if (base >= total) {
        // Entire quad beyond total: pass through m_indices (vector NT load).
        __builtin_prefetch(m_indices + base, 0, 0);      // global_prefetch_b8
        const v4i mi =
            __builtin_nontemporal_load((const v4i*)(m_indices + base));
        v = (v4f){(float)mi.x, (float)mi.y, (float)mi.z, (float)mi.w};
    } else {
        // Straddling quad (at most one in the whole grid): per-element.
#pragma unroll
        for (int j = 0; j < 4; ++j) {
            const int idx = base + j;
            float val;
            if (idx < total) {
                int e = 0;
#pragma unroll
                for (int step = 32; step >= 1; step >>= 1) {
                    if (s_starts[e + step] <= idx) e += step;
                }
                val = (float)e;
            } else {
                val = (float)m_indices[idx];
            }
            v[j] = val;
        }
    }
    __builtin_nontemporal_store(v, (v4f*)(out + base));  // global_store_b128 NT
}

// ---------------------------------------------------------------------------
// Launch: inputs are {num_recv_tokens_per_expert, expert_start_loc, m_indices}.
// d_out = [m_out (TOTAL_TOKENS floats) | starts (64 floats)].
// ---------------------------------------------------------------------------
extern "C" void kernel_launch(void* const* d_in, const int* in_sizes, int n_in,
                              void* d_out, int out_size, void* d_ws, size_t ws_size,
                              hipStream_t stream) {
    const int* counts    = (const int*)d_in[0];
    const int* m_indices = (const int*)d_in[2];
    float*     out       = (float*)d_out;
    int*       ws        = (int*)d_ws;   // [65] ints: starts + total

    expert_scan_kernel<<<1, N_EXPERTS, 0, stream>>>(counts, ws, out + TOTAL_TOKENS);

    const int quads  = TOTAL_TOKENS / 4;       // 16,777,216
    const int block  = 256;
    const int grid   = quads / block;          // 65,536 blocks
    fill_m_indices_kernel<<<grid, block, 0, stream>>>(ws, m_indices, out);
}